// HamModule_54872502174449
// MI455X (gfx1250) — compile-verified
//
#include <hip/hip_runtime.h>
#include <math.h>

// ---------------------------------------------------------------------------
// Problem constants (match reference)
// ---------------------------------------------------------------------------
#define LHALF 6
#define NSITE 12
#define DDIM  4096          // 2^12
#define KSUB  16            // iteration block size (one WMMA tile wide)
#define NEIG  4
#define J1    1.0f
#define SIGMA 32.0f         // > Gershgorin bound on |lambda(H)| (<= 27)
#define PI_F  3.14159265358979323846f

// Workspace layout (float offsets into d_ws)
#define BZ_OFF   0                         // Bz[12]
#define BX_OFF   16                        // Bx[12]
#define EV_OFF   32                        // eigenvalues (16, use 4)
#define DIAG_OFF 64                        // diag[4096]
#define G_OFF    4160                      // 16x16 Gram / Rayleigh matrix
#define L_OFF    4416                      // 16x16 Cholesky factor
#define M_OFF    4672                      // 16x16 L^{-T} (dense, upper-tri)
#define Y_OFF    4928                      // 16x16 Ritz rotation (cols 4..15 = 0)
#define V_OFF    5184                      // 4096x16 subspace
#define W_OFF    (V_OFF + DDIM*KSUB)       // 4096x16 scratch
#define GP_OFF   (W_OFF + DDIM*KSUB)       // 64 partial 16x16 tiles
#define NGRAMW   64                        // gram waves (one per block)

typedef float v2f __attribute__((ext_vector_type(2)));
typedef float v4f __attribute__((ext_vector_type(4)));
typedef float v8f __attribute__((ext_vector_type(8)));

// ---------------------------------------------------------------------------
// 1. Field setup: site angles -> Bz[i], Bx[i]  (single thread, trivial)
// ---------------------------------------------------------------------------
__global__ void k_fields(const float* __restrict__ B0,
                         const float* __restrict__ Bext,
                         const float* __restrict__ phi_i,
                         float* __restrict__ ws) {
    if (threadIdx.x != 0 || blockIdx.x != 0) return;
    float c[LHALF];
    float run = 0.0f;
    for (int k = 0; k < LHALF; ++k) { run += phi_i[k] * phi_i[k]; c[k] = run; }
    float scale = PI_F / run;
    float base = c[0] * scale;
    float ang[NSITE];
    for (int k = 0; k < LHALF; ++k) ang[k] = c[k] * scale - base;
    for (int k = 0; k < LHALF; ++k) ang[LHALF + k] = 2.0f * PI_F - (c[LHALF - 1 - k] * scale - base);
    float b0 = B0[0], be = Bext[0];
    for (int n = 0; n < NSITE; ++n) {
        ws[BZ_OFF + n] = b0 * cosf(ang[n]) + be;
        ws[BX_OFF + n] = b0 * sinf(ang[n]);
    }
}

// ---------------------------------------------------------------------------
// 2. Diagonal of H per basis state:  J1*sum Sz_i Sz_{i+1}  +  sum Sz_i Bz_i
// ---------------------------------------------------------------------------
__global__ __launch_bounds__(256) void k_diag(const float* __restrict__ ws,
                                              float* __restrict__ diag) {
    int s = blockIdx.x * 256 + threadIdx.x;
    if (s >= DDIM) return;
    float acc = 0.0f;
#pragma unroll
    for (int i = 0; i < NSITE; ++i) {
        float szi = ((s >> i) & 1) ? 0.5f : -0.5f;
        int j = (i + 1) % NSITE;
        float szj = ((s >> j) & 1) ? 0.5f : -0.5f;
        acc += J1 * szi * szj;
        acc += szi * ws[BZ_OFF + i];
    }
    diag[s] = acc;
}

// ---------------------------------------------------------------------------
// 3. Deterministic pseudo-random init of the K=16 subspace
// ---------------------------------------------------------------------------
__global__ __launch_bounds__(256) void k_init(float* __restrict__ V) {
    int idx = blockIdx.x * 256 + threadIdx.x;     // 0 .. 65535
    if (idx >= DDIM * KSUB) return;
    unsigned h = (unsigned)idx * 2654435761u;
    h ^= h >> 16; h *= 2246822519u; h ^= h >> 13; h *= 3266489917u; h ^= h >> 16;
    V[idx] = ((float)(h & 0xFFFFFF) * (1.0f / 16777216.0f)) - 0.5f;
}

// ---------------------------------------------------------------------------
// 4. Matrix-free block SpMV:  Y[s][:] = coefH*(H X)[s][:] + coefI*X[s][:]
//    25 structured nnz/row (diag + 12 single-flips + <=12 exchange), all
//    XOR-indexed.  float4 ext-vectors force b128 traffic; the 256KB panels
//    stay resident in L2 (192MB) so this is latency-, not HBM-, bound.
// ---------------------------------------------------------------------------
__global__ __launch_bounds__(256) void k_spmv(const float* __restrict__ X,
                                              float* __restrict__ Y,
                                              const float* __restrict__ ws,
                                              const float* __restrict__ diag,
                                              float coefH, float coefI) {
    int s = blockIdx.x * 256 + threadIdx.x;
    if (s >= DDIM) return;
    const v4f* xs = (const v4f*)(X + (size_t)s * KSUB);
    v4f x0 = xs[0], x1 = xs[1], x2 = xs[2], x3 = xs[3];
    float d = diag[s];
    v4f a0 = d * x0, a1 = d * x1, a2 = d * x2, a3 = d * x3;
#pragma unroll
    for (int i = 0; i < NSITE; ++i) {
        float bx = 0.5f * ws[BX_OFF + i];
        const v4f* xt = (const v4f*)(X + (size_t)(s ^ (1 << i)) * KSUB);
        a0 += bx * xt[0]; a1 += bx * xt[1]; a2 += bx * xt[2]; a3 += bx * xt[3];
        const int j = (i + 1) % NSITE;
        if ((((s >> i) ^ (s >> j)) & 1) != 0) {
            const v4f* xe = (const v4f*)(X + (size_t)(s ^ ((1 << i) | (1 << j))) * KSUB);
            a0 += (0.5f * J1) * xe[0]; a1 += (0.5f * J1) * xe[1];
            a2 += (0.5f * J1) * xe[2]; a3 += (0.5f * J1) * xe[3];
        }
    }
    v4f* ys = (v4f*)(Y + (size_t)s * KSUB);
    ys[0] = coefH * a0 + coefI * x0;
    ys[1] = coefH * a1 + coefI * x1;
    ys[2] = coefH * a2 + coefI * x2;
    ys[3] = coefH * a3 + coefI * x3;
}

// ---------------------------------------------------------------------------
// 5. Gram projection G = A^T B (16x16) over 4096 rows with fp32 WMMA
//    (16x16x4).  Each wave owns a contiguous span of 16 row-chunks; x4 unroll
//    lets the compiler clause 16 loads ahead of 4 back-to-back v_wmma ops.
//    fp32 A-frag layout (ISA 7.12.2): lanes 0-15 hold M=lane with K0/K1,
//    lanes 16-31 hold K2/K3.  C/D: VGPR v -> row v + 8*(lane>=16), col lane&15.
//    Per-wave partial tiles + fixed-order reduction => bitwise deterministic.
// ---------------------------------------------------------------------------
__global__ __launch_bounds__(32) void k_gram(const float* __restrict__ A,
                                             const float* __restrict__ B,
                                             float* __restrict__ Gpart) {
    const int lane  = threadIdx.x;           // one wave per block, EXEC all ones
    const int wid   = blockIdx.x;            // 0 .. NGRAMW-1
    const int m     = lane & 15;
    const int khalf = (lane >> 4) * 2;       // 0 for lanes 0-15, 2 for lanes 16-31
    v8f c = {0.f, 0.f, 0.f, 0.f, 0.f, 0.f, 0.f, 0.f};
    const int per = (DDIM / 4) / NGRAMW;     // 16 chunks of 4 rows per wave
    const int c0  = wid * per;
#pragma unroll 4
    for (int chunk = c0; chunk < c0 + per; ++chunk) {
        const int rb = chunk * 4;
        v2f a, b;
        a.x = A[(size_t)(rb + khalf)     * KSUB + m];
        a.y = A[(size_t)(rb + khalf + 1) * KSUB + m];
        b.x = B[(size_t)(rb + khalf)     * KSUB + m];
        b.y = B[(size_t)(rb + khalf + 1) * KSUB + m];
        c = __builtin_amdgcn_wmma_f32_16x16x4_f32(false, a, false, b,
                                                  (short)0, c, false, false);
    }
    const int mbase = (lane >> 4) ? 8 : 0;
#pragma unroll
    for (int v = 0; v < 8; ++v)
        Gpart[(size_t)wid * 256 + (size_t)(v + mbase) * 16 + m] = c[v];
}

__global__ __launch_bounds__(256) void k_gred(const float* __restrict__ Gpart,
                                              float* __restrict__ G) {
    int t = threadIdx.x;                     // 0..255, one Gram element each
    float acc = 0.0f;
    for (int w = 0; w < NGRAMW; ++w) acc += Gpart[(size_t)w * 256 + t];
    G[t] = acc;
}

// ---------------------------------------------------------------------------
// 6. WMMA right-multiply: O[rb:rb+16][:] = X[rb:rb+16][:] * Bm (16x16).
//    One wave per 16-row tile, 4 k-steps of 16x16x4.  A-frag loads are
//    aligned b64 (each lane reads 2 consecutive k of its own row).
//    Used for the orthonormalization (Bm = L^{-T}) and the final Ritz
//    projection (Bm = Y zero-padded).  In-place safe: tiles are disjoint.
// ---------------------------------------------------------------------------
__global__ __launch_bounds__(32) void k_gemm(const float* __restrict__ X,
                                             float* __restrict__ O,
                                             const float* __restrict__ Bm) {
    const int lane  = threadIdx.x;           // EXEC all ones
    const int rb    = blockIdx.x * 16;       // 256 blocks cover 4096 rows
    const int n     = lane & 15;
    const int khalf = (lane >> 4) * 2;
    // Preload B fragments for the 4 k-steps (k0 = 0,4,8,12)
    v2f bf[4];
#pragma unroll
    for (int t = 0; t < 4; ++t) {
        bf[t].x = Bm[(size_t)(4 * t + khalf)     * KSUB + n];
        bf[t].y = Bm[(size_t)(4 * t + khalf + 1) * KSUB + n];
    }
    v8f c = {0.f, 0.f, 0.f, 0.f, 0.f, 0.f, 0.f, 0.f};
#pragma unroll
    for (int t = 0; t < 4; ++t) {
        // A[m][k] = X[rb+m][4t+k]; lane reads 2 consecutive floats of its row
        v2f a = *(const v2f*)(X + (size_t)(rb + n) * KSUB + 4 * t + khalf);
        c = __builtin_amdgcn_wmma_f32_16x16x4_f32(false, a, false, bf[t],
                                                  (short)0, c, false, false);
    }
    const int mbase = (lane >> 4) ? 8 : 0;
#pragma unroll
    for (int v = 0; v < 8; ++v)
        O[(size_t)(rb + v + mbase) * KSUB + n] = c[v];
}

// ---------------------------------------------------------------------------
// 7. 16x16 Cholesky (ridge) + triangular inverse -> dense Minv = L^{-T}
//    Single thread; negligible work.
// ---------------------------------------------------------------------------
__global__ void k_chol(const float* __restrict__ G, float* __restrict__ Lm,
                       float* __restrict__ Minv) {
    if (threadIdx.x != 0 || blockIdx.x != 0) return;
    float tr = 0.0f;
    for (int i = 0; i < 16; ++i) tr += G[i * 17];
    float ridge = 1e-6f * tr + 1e-20f;
    for (int j = 0; j < 16; ++j) {
        float d = G[j * 16 + j] + ridge;
        for (int k = 0; k < j; ++k) d -= Lm[j * 16 + k] * Lm[j * 16 + k];
        d = fmaxf(d, 1e-20f);
        float Ljj = sqrtf(d);
        Lm[j * 16 + j] = Ljj;
        float inv = 1.0f / Ljj;
        for (int i = j + 1; i < 16; ++i) {
            float sum = G[i * 16 + j];
            for (int k = 0; k < j; ++k) sum -= Lm[i * 16 + k] * Lm[j * 16 + k];
            Lm[i * 16 + j] = sum * inv;
        }
        for (int i = 0; i < j; ++i) Lm[i * 16 + j] = 0.0f;   // zero upper part
    }
    // Linv = L^{-1} (lower triangular), then Minv = Linv^T (upper triangular)
    float Linv[16][16];
    for (int i = 0; i < 16; ++i)
        for (int j = 0; j < 16; ++j) Linv[i][j] = 0.0f;
    for (int i = 0; i < 16; ++i) {
        Linv[i][i] = 1.0f / Lm[i * 16 + i];
        for (int j = 0; j < i; ++j) {
            float sum = 0.0f;
            for (int k = j; k < i; ++k) sum += Lm[i * 16 + k] * Linv[k][j];
            Linv[i][j] = -sum / Lm[i * 16 + i];
        }
    }
    for (int k = 0; k < 16; ++k)
        for (int nn = 0; nn < 16; ++nn)
            Minv[k * 16 + nn] = Linv[nn][k];                 // (L^{-1})^T
}

// ---------------------------------------------------------------------------
// 8. 16x16 symmetric Jacobi eigensolve (Rayleigh-Ritz) -- single thread.
//    Writes eigvals[0:4] and a zero-padded 16x16 rotation (cols 4..15 = 0)
//    so the Ritz projection can reuse the WMMA GEMM.
// ---------------------------------------------------------------------------
__global__ void k_eig16(const float* __restrict__ S,
                        float* __restrict__ ev, float* __restrict__ Ypad) {
    if (threadIdx.x != 0 || blockIdx.x != 0) return;
    float a[16][16], q[16][16];
    for (int i = 0; i < 16; ++i)
        for (int j = 0; j < 16; ++j) {
            a[i][j] = 0.5f * (S[i * 16 + j] + S[j * 16 + i]);
            q[i][j] = (i == j) ? 1.0f : 0.0f;
        }
    for (int sweep = 0; sweep < 30; ++sweep) {
        for (int p = 0; p < 15; ++p)
            for (int r = p + 1; r < 16; ++r) {
                float apq = a[p][r];
                if (fabsf(apq) < 1e-12f) continue;
                float theta = 0.5f * (a[r][r] - a[p][p]) / apq;
                float t = (theta >= 0.0f ? 1.0f : -1.0f) /
                          (fabsf(theta) + sqrtf(1.0f + theta * theta));
                float cc = 1.0f / sqrtf(1.0f + t * t);
                float ss = t * cc;
                for (int k = 0; k < 16; ++k) {
                    float akp = a[k][p], akr = a[k][r];
                    a[k][p] = cc * akp - ss * akr;
                    a[k][r] = ss * akp + cc * akr;
                }
                for (int k = 0; k < 16; ++k) {
                    float apk = a[p][k], ark = a[r][k];
                    a[p][k] = cc * apk - ss * ark;
                    a[r][k] = ss * apk + cc * ark;
                }
                for (int k = 0; k < 16; ++k) {
                    float qkp = q[k][p], qkr = q[k][r];
                    q[k][p] = cc * qkp - ss * qkr;
                    q[k][r] = ss * qkp + cc * qkr;
                }
            }
    }
    int idx[16];
    for (int i = 0; i < 16; ++i) idx[i] = i;
    for (int i = 0; i < 16; ++i) {             // selection sort, ascending
        int m = i;
        for (int j = i + 1; j < 16; ++j)
            if (a[idx[j]][idx[j]] < a[idx[m]][idx[m]]) m = j;
        int tmp = idx[i]; idx[i] = idx[m]; idx[m] = tmp;
    }
    for (int j = 0; j < NEIG; ++j) ev[j] = a[idx[j]][idx[j]];
    for (int k = 0; k < 16; ++k)
        for (int j = 0; j < 16; ++j)
            Ypad[k * 16 + j] = (j < NEIG) ? q[k][idx[j]] : 0.0f;
}

// ---------------------------------------------------------------------------
// 9. Pack output: row 0 = eigvals[0:4]; rows 1..4096 = P[s][0:4]
//    (P = V*Ypad, already computed by the WMMA GEMM into W)
// ---------------------------------------------------------------------------
__global__ __launch_bounds__(256) void k_out(const float* __restrict__ P,
                                             const float* __restrict__ ev,
                                             float* __restrict__ out) {
    int idx = blockIdx.x * 256 + threadIdx.x;
    if (idx > DDIM) return;
    if (idx == 0) {
#pragma unroll
        for (int j = 0; j < NEIG; ++j) out[j] = ev[j];
        return;
    }
    int s = idx - 1;
#pragma unroll
    for (int j = 0; j < NEIG; ++j)
        out[(size_t)idx * NEIG + j] = P[(size_t)s * KSUB + j];
}

// ---------------------------------------------------------------------------
// Host orchestration (graph-capture safe: kernel launches only)
// ---------------------------------------------------------------------------
extern "C" void kernel_launch(void* const* d_in, const int* in_sizes, int n_in,
                              void* d_out, int out_size, void* d_ws, size_t ws_size,
                              hipStream_t stream) {
    (void)in_sizes; (void)n_in; (void)out_size; (void)ws_size;
    const float* B0   = (const float*)d_in[0];
    const float* Bext = (const float*)d_in[1];
    const float* phi  = (const float*)d_in[2];
    float* ws   = (float*)d_ws;
    float* out  = (float*)d_out;
    float* diag = ws + DIAG_OFF;
    float* G    = ws + G_OFF;
    float* Lm   = ws + L_OFF;
    float* Mi   = ws + M_OFF;
    float* Yp   = ws + Y_OFF;
    float* ev   = ws + EV_OFF;
    float* V    = ws + V_OFF;
    float* W    = ws + W_OFF;
    float* Gp   = ws + GP_OFF;

    k_fields<<<1, 1, 0, stream>>>(B0, Bext, phi, ws);
    k_diag  <<<DDIM / 256, 256, 0, stream>>>(ws, diag);
    k_init  <<<DDIM * KSUB / 256, 256, 0, stream>>>(V);

    // Block subspace iteration on A = SIGMA*I - H (SPD; dominant subspace of A
    // == lowest subspace of H).  8 SpMVs per orthonormalization, 40 rounds.
    for (int it = 0; it < 40; ++it) {
        for (int p = 0; p < 4; ++p) {
            k_spmv<<<DDIM / 256, 256, 0, stream>>>(V, W, ws, diag, -1.0f, SIGMA);
            k_spmv<<<DDIM / 256, 256, 0, stream>>>(W, V, ws, diag, -1.0f, SIGMA);
        }
        k_gram <<<NGRAMW, 32, 0, stream>>>(V, V, Gp);      // G = V^T V (WMMA)
        k_gred <<<1, 256, 0, stream>>>(Gp, G);
        k_chol <<<1, 1, 0, stream>>>(G, Lm, Mi);           // Mi = L^{-T}
        k_gemm <<<DDIM / 16, 32, 0, stream>>>(V, V, Mi);   // V = V*L^{-T} (WMMA)
    }

    // Rayleigh-Ritz on H itself: S = V^T (H V), small eigensolve, project out.
    k_spmv <<<DDIM / 256, 256, 0, stream>>>(V, W, ws, diag, 1.0f, 0.0f);
    k_gram <<<NGRAMW, 32, 0, stream>>>(V, W, Gp);          // S = V^T W (WMMA)
    k_gred <<<1, 256, 0, stream>>>(Gp, G);
    k_eig16<<<1, 1, 0, stream>>>(G, ev, Yp);
    k_gemm <<<DDIM / 16, 32, 0, stream>>>(V, W, Yp);       // P = V*Ypad (WMMA)
    k_out  <<<(DDIM + 256) / 256 + 1, 256, 0, stream>>>(W, ev, out);
}